// NetworkLocal_53102975648382
// MI455X (gfx1250) — compile-verified
//
#include <hip/hip_runtime.h>
#include <hip/hip_bf16.h>

typedef float v2f __attribute__((ext_vector_type(2)));
typedef float v8f __attribute__((ext_vector_type(8)));

#define EPS 1e-5f
#define LRELU_SLOPE 0.01f

// Fixed problem dims from the reference
#define BB   2
#define TT   1024
#define NDIM 256      // ND
#define KB   3
#define KIN  19
#define KPAD 20       // pad K to multiple of 4
#define HH   128
#define LL   128
#define BT   (BB * TT)

// slope < 1  =>  leaky_relu(x) == max(x, slope*x); lowers to v_mul + v_max_num_f32
__device__ __forceinline__ float lrelu(float v) {
    return fmaxf(v, LRELU_SLOPE * v);
}

// -----------------------------------------------------------------------------
// Kernel 1: one workgroup per (b,t). Builds the 19 (padded to 20) scalar
// features for ND=256 points, runs [256x20]x[20x128] with v_wmma_f32_16x16x4_f32,
// fuses bias + leaky_relu + mean-pool over ND, writes H1[bt][128] to workspace.
// -----------------------------------------------------------------------------
__global__ __launch_bounds__(128) void layer0_kernel(
    const float* __restrict__ x0g,    // [B,T,3]
    const float* __restrict__ xg,     // [B,T,ND,3]
    const int*   __restrict__ Ng,     // [B,T]
    const float* __restrict__ basisg, // [B,KB,3]
    const float* __restrict__ vg,     // [B,T,ND,3]
    const float* __restrict__ W0g,    // [19,127]
    const float* __restrict__ b0g,    // [127]
    float* __restrict__ H1)           // [B*T,128] workspace
{
    __shared__ float S[NDIM][KPAD];     // scalar features, K padded
    __shared__ float W0p[KPAD][HH];     // padded W0 (col 127 / rows 19 zero)
    __shared__ float b0p[HH];
    __shared__ float colsum[HH];
    __shared__ float red[3][128];

    const int tid  = threadIdx.x;
    const int bt   = blockIdx.x;
    const int b    = bt >> 10;          // T = 1024
    const long pbase = (long)bt * NDIM;

    // ---- basis rows (broadcast loads) ----
    float bas[3][3];
    #pragma unroll
    for (int k = 0; k < KB; ++k)
        #pragma unroll
        for (int c = 0; c < 3; ++c)
            bas[k][c] = basisg[(b * KB + k) * 3 + c];

    // ---- v0 = mean over ND of v ----
    float sx = 0.f, sy = 0.f, sz = 0.f;
    for (int n = tid; n < NDIM; n += 128) {
        const float* vp = vg + (pbase + n) * 3;
        sx += vp[0]; sy += vp[1]; sz += vp[2];
    }
    red[0][tid] = sx; red[1][tid] = sy; red[2][tid] = sz;
    __syncthreads();
    for (int s = 64; s > 0; s >>= 1) {
        if (tid < s) {
            red[0][tid] += red[0][tid + s];
            red[1][tid] += red[1][tid + s];
            red[2][tid] += red[2][tid + s];
        }
        __syncthreads();
    }
    const float v0x = red[0][0] * (1.f / NDIM);
    const float v0y = red[1][0] * (1.f / NDIM);
    const float v0z = red[2][0] * (1.f / NDIM);

    // ---- per-(b,t) constant features (computed redundantly per thread) ----
    const float x0x = x0g[bt * 3 + 0], x0y = x0g[bt * 3 + 1], x0z = x0g[bt * 3 + 2];
    const float x0n = sqrtf(x0x * x0x + x0y * x0y + x0z * x0z) + EPS;
    const float x0ux = x0x / x0n, x0uy = x0y / x0n, x0uz = x0z / x0n;
    const float v0n = sqrtf(v0x * v0x + v0y * v0y + v0z * v0z) + EPS;
    const float v0ux = v0x / v0n, v0uy = v0y / v0n, v0uz = v0z / v0n;
    const float x0v0 = x0ux * v0ux + x0uy * v0uy + x0uz * v0uz;
    const float logN = log1pf((float)Ng[bt]);
    float x0b[3], v0b[3];
    #pragma unroll
    for (int k = 0; k < KB; ++k) {
        x0b[k] = x0ux * bas[k][0] + x0uy * bas[k][1] + x0uz * bas[k][2];
        v0b[k] = v0ux * bas[k][0] + v0uy * bas[k][1] + v0uz * bas[k][2];
    }

    // ---- per-point features into LDS (2 points per thread) ----
    for (int n = tid; n < NDIM; n += 128) {
        const float* xp = xg + (pbase + n) * 3;
        const float* vp = vg + (pbase + n) * 3;
        float xx = xp[0], xy = xp[1], xz = xp[2];
        float vx = vp[0] - v0x, vy = vp[1] - v0y, vz = vp[2] - v0z;
        float xn = sqrtf(xx * xx + xy * xy + xz * xz) + EPS;
        float rxn = 1.f / xn;
        float xux = xx * rxn, xuy = xy * rxn, xuz = xz * rxn;
        float vn = sqrtf(vx * vx + vy * vy + vz * vz) + EPS;
        float rvn = 1.f / vn;
        float vux = vx * rvn, vuy = vy * rvn, vuz = vz * rvn;

        float* Sr = S[n];
        Sr[0] = xn;
        #pragma unroll
        for (int k = 0; k < KB; ++k)
            Sr[1 + k] = xux * bas[k][0] + xuy * bas[k][1] + xuz * bas[k][2];
        Sr[4] = logN;
        Sr[5] = x0n;
        Sr[6] = x0b[0]; Sr[7] = x0b[1]; Sr[8] = x0b[2];
        Sr[9] = v0n;
        Sr[10] = v0b[0]; Sr[11] = v0b[1]; Sr[12] = v0b[2];
        Sr[13] = x0v0;
        Sr[14] = vn;
        #pragma unroll
        for (int k = 0; k < KB; ++k)
            Sr[15 + k] = vux * bas[k][0] + vuy * bas[k][1] + vuz * bas[k][2];
        Sr[18] = xux * vux + xuy * vuy + xuz * vuz;
        Sr[19] = 0.f;                          // K padding column
    }

    // ---- padded W0 / b0 into LDS, zero colsum ----
    for (int idx = tid; idx < KPAD * HH; idx += 128) {
        int k = idx >> 7, j = idx & 127;
        W0p[k][j] = (k < KIN && j < (HH - 1)) ? W0g[k * (HH - 1) + j] : 0.f;
    }
    b0p[tid]    = (tid < HH - 1) ? b0g[tid] : 0.f;
    colsum[tid] = 0.f;
    __syncthreads();

    // ---- WMMA: [256x20] x [20x128], fused bias+lrelu+column-sum ----
    const int lane = tid & 31;
    const int wv   = tid >> 5;          // wave id (wave32)
    const int half = lane >> 4;         // 0: lanes 0-15, 1: lanes 16-31
    const int l15  = lane & 15;

    for (int nt = 0; nt < 8; ++nt) {
        const int col = nt * 16 + l15;
        const float bcol = b0p[col];
        float csum = 0.f;
        #pragma unroll
        for (int mt = 0; mt < 4; ++mt) {
            const int row = (wv * 4 + mt) * 16 + l15;
            v8f c = {};
            #pragma unroll
            for (int kk = 0; kk < 5; ++kk) {
                const int k0 = kk * 4 + half * 2;   // K slot for this lane half
                v2f a, bb;
                a.x  = S[row][k0];
                a.y  = S[row][k0 + 1];
                bb.x = W0p[k0][col];
                bb.y = W0p[k0 + 1][col];
                c = __builtin_amdgcn_wmma_f32_16x16x4_f32(
                        false, a, false, bb, (short)0, c, false, false);
            }
            #pragma unroll
            for (int i = 0; i < 8; ++i)
                csum += lrelu(c[i] + bcol);
        }
        atomicAdd(&colsum[col], csum);   // ds_add_f32; lanes l and l+16 share col
    }
    __syncthreads();

    // ---- mean over ND, concat log1p(N) at col 127 ----
    H1[(long)bt * HH + tid] = (tid == HH - 1) ? logN : colsum[tid] * (1.f / NDIM);
}

// -----------------------------------------------------------------------------
// Kernel 2: rows of H1 -> leaky_relu(H1*W1+b1) -> (*W2+b2) * (1/P200c[b]).
// 32 rows per workgroup, 4 waves; two WMMA GEMMs with K=128 (32 k-steps).
// B-fragment loaded once per k-step, shared by both M-tile accumulators.
// -----------------------------------------------------------------------------
__global__ __launch_bounds__(128) void mlp_kernel(
    const float* __restrict__ H1,     // [B*T,128]
    const float* __restrict__ W1g,    // [128,128]
    const float* __restrict__ b1g,    // [128]
    const float* __restrict__ W2g,    // [128,128]
    const float* __restrict__ b2g,    // [128]
    const float* __restrict__ P200c,  // [B]
    float* __restrict__ out)          // [B*T,128]
{
    __shared__ float Ht[32][HH];
    __shared__ float Gt[32][HH];

    const int tid  = threadIdx.x;
    const int row0 = blockIdx.x * 32;
    const int lane = tid & 31;
    const int wv   = tid >> 5;
    const int half = lane >> 4;
    const int l15  = lane & 15;

    const float invP = 1.f / P200c[row0 >> 10];   // whole WG shares one b

    for (int idx = tid; idx < 32 * HH; idx += 128) {
        int r = idx >> 7, j = idx & 127;
        Ht[r][j] = H1[(long)(row0 + r) * HH + j];
    }
    __syncthreads();

    // ---- layer 1: Gt = lrelu(Ht * W1 + b1) ----
    #pragma unroll
    for (int ntl = 0; ntl < 2; ++ntl) {
        const int n0  = (wv * 2 + ntl) * 16;
        const int col = n0 + l15;
        const float bcol = b1g[col];
        v8f c0 = {}, c1 = {};
        const int rowA = 0  + l15;
        const int rowB = 16 + l15;
        for (int kk = 0; kk < 32; ++kk) {
            const int k0 = kk * 4 + half * 2;
            v2f bb, a0, a1;
            bb.x = W1g[k0 * HH + col];
            bb.y = W1g[(k0 + 1) * HH + col];
            a0.x = Ht[rowA][k0];  a0.y = Ht[rowA][k0 + 1];
            a1.x = Ht[rowB][k0];  a1.y = Ht[rowB][k0 + 1];
            c0 = __builtin_amdgcn_wmma_f32_16x16x4_f32(
                     false, a0, false, bb, (short)0, c0, false, false);
            c1 = __builtin_amdgcn_wmma_f32_16x16x4_f32(
                     false, a1, false, bb, (short)0, c1, false, false);
        }
        #pragma unroll
        for (int i = 0; i < 8; ++i) {
            Gt[i + 8 * half][col]      = lrelu(c0[i] + bcol);
            Gt[16 + i + 8 * half][col] = lrelu(c1[i] + bcol);
        }
    }
    __syncthreads();

    // ---- layer 2: out = (Gt * W2 + b2) * invP ----
    #pragma unroll
    for (int ntl = 0; ntl < 2; ++ntl) {
        const int n0  = (wv * 2 + ntl) * 16;
        const int col = n0 + l15;
        const float bcol = b2g[col];
        v8f c0 = {}, c1 = {};
        const int rowA = 0  + l15;
        const int rowB = 16 + l15;
        for (int kk = 0; kk < 32; ++kk) {
            const int k0 = kk * 4 + half * 2;
            v2f bb, a0, a1;
            bb.x = W2g[k0 * HH + col];
            bb.y = W2g[(k0 + 1) * HH + col];
            a0.x = Gt[rowA][k0];  a0.y = Gt[rowA][k0 + 1];
            a1.x = Gt[rowB][k0];  a1.y = Gt[rowB][k0 + 1];
            c0 = __builtin_amdgcn_wmma_f32_16x16x4_f32(
                     false, a0, false, bb, (short)0, c0, false, false);
            c1 = __builtin_amdgcn_wmma_f32_16x16x4_f32(
                     false, a1, false, bb, (short)0, c1, false, false);
        }
        #pragma unroll
        for (int i = 0; i < 8; ++i) {
            const long rA = row0 + i + 8 * half;
            const long rB = row0 + 16 + i + 8 * half;
            out[rA * LL + col] = (c0[i] + bcol) * invP;
            out[rB * LL + col] = (c1[i] + bcol) * invP;
        }
    }
}

extern "C" void kernel_launch(void* const* d_in, const int* in_sizes, int n_in,
                              void* d_out, int out_size, void* d_ws, size_t ws_size,
                              hipStream_t stream) {
    const float* x0    = (const float*)d_in[0];
    const float* x     = (const float*)d_in[1];
    const int*   N     = (const int*)  d_in[2];
    const float* basis = (const float*)d_in[3];
    const float* v     = (const float*)d_in[4];
    const float* P200c = (const float*)d_in[5];
    const float* W0    = (const float*)d_in[6];
    const float* b0    = (const float*)d_in[7];
    const float* W1    = (const float*)d_in[8];
    const float* b1    = (const float*)d_in[9];
    const float* W2    = (const float*)d_in[10];
    const float* b2    = (const float*)d_in[11];
    float* out = (float*)d_out;
    float* H1  = (float*)d_ws;   // needs B*T*128*4 = 1 MB

    layer0_kernel<<<BT, 128, 0, stream>>>(x0, x, N, basis, v, W0, b0, H1);
    mlp_kernel<<<BT / 32, 128, 0, stream>>>(H1, W1, b1, W2, b2, P200c, out);
}